// MainModel_16088947491524
// MI455X (gfx1250) — compile-verified
//
#include <hip/hip_runtime.h>
#include <math.h>

// Problem constants (from reference)
#define C_    512
#define H_    14
#define W_    14
#define HW    196           // 14*14
#define P_    49
#define KTOT  50176         // 512 * 2 * 49  (x_cat_ columns)
#define NCLS  1000
#define NPAD  1008          // 63 * 16
#define NB    63            // N blocks of 16
#define KSPLIT 14           // K split factor (blockIdx.y)
#define KTILE 256           // LDS K tile
#define LSTR  260           // padded LDS row stride (260 % 64 == 4 -> conflict-free)
#define TILES_PER_WG 14     // 50176 / 256 / 14

typedef __attribute__((ext_vector_type(2))) float v2f;
typedef __attribute__((ext_vector_type(8))) float v8f;
typedef unsigned int u32x4 __attribute__((ext_vector_type(4)));
typedef int          i32x4 __attribute__((ext_vector_type(4)));
typedef int          i32x8 __attribute__((ext_vector_type(8)));

#if defined(__gfx1250__) && __has_builtin(__builtin_amdgcn_tensor_load_to_lds) && \
    __has_builtin(__builtin_amdgcn_s_wait_tensorcnt)
#define USE_TDM 1
#else
#define USE_TDM 0
#endif

// ---------------------------------------------------------------------------
// K1a: x_avepool = max over (H,W) -> [10,512]
__global__ void reduce_max_kernel(const float* __restrict__ x, float* __restrict__ xave) {
    int gid = blockIdx.x * blockDim.x + threadIdx.x;
    if (gid >= 10 * C_) return;
    const float* px = x + (size_t)gid * HW;
    float m = px[0];
    for (int i = 1; i < HW; ++i) m = fmaxf(m, px[i]);
    xave[gid] = m;
}

// ---------------------------------------------------------------------------
// K1m: mask = tanh(avgpool2(1x1conv(x))) -> [10,49], written to d_out[25000..)
__global__ void mask_kernel(const float* __restrict__ x, const float* __restrict__ cw,
                            const float* __restrict__ cb, float* __restrict__ out) {
    int gid = blockIdx.x * blockDim.x + threadIdx.x;
    if (gid >= 10 * P_) return;
    int b = gid / P_, p = gid % P_;
    int h0 = 2 * (p / 7), w0 = 2 * (p % 7);
    const float* xb = x + (size_t)b * C_ * HW;
    float s = 0.f;
    for (int c = 0; c < C_; ++c) {
        const float* xc_ = xb + c * HW;
        float t = xc_[h0 * W_ + w0] + xc_[h0 * W_ + w0 + 1]
                + xc_[(h0 + 1) * W_ + w0] + xc_[(h0 + 1) * W_ + w0 + 1];
        s += t * cw[c];
    }
    out[25000 + gid] = tanhf(0.25f * s + cb[0]);
}

// ---------------------------------------------------------------------------
// K1b: build x_cat_ [5, 50176] directly from x (fused avgpool + patch permute)
__global__ void build_xcat_kernel(const float* __restrict__ x, const int* __restrict__ idx,
                                  float* __restrict__ xc) {
    int gid = blockIdx.x * blockDim.x + threadIdx.x;
    if (gid >= 5 * KTOT) return;
    int s = gid / KTOT, k = gid % KTOT;
    int b, c, h, w;
    if (k < KTOT / 2) {                       // unswap half: flat (c,h,w) of xp[2s]
        b = 2 * s; c = k / P_; int p = k % P_; h = p / 7; w = p % 7;
    } else {                                  // reordered half: memory-reinterp gather
        int q = k - KTOT / 2;
        b = 2 * s + 1;
        int r = q / P_, j = q % P_;
        int c2 = idx[b * P_ + j];             // index_dy[2s+1][j]
        int f = r * P_ + c2;                  // flat in (w,c,h) order, stride 3584
        w = f / (C_ * 7);
        int rem = f % (C_ * 7);
        c = rem / 7; h = rem % 7;
    }
    const float* px = x + ((size_t)b * C_ + c) * HW + (2 * h) * W_ + 2 * w;
    xc[gid] = 0.25f * (px[0] + px[1] + px[W_] + px[W_ + 1]);
}

// ---------------------------------------------------------------------------
// K2: out_swap = x_avepool @ classifier_swap_w.T -> [10,2000], one block per class j
__global__ __launch_bounds__(256) void swap_gemm_kernel(const float* __restrict__ xave,
                                                        const float* __restrict__ Wsw,
                                                        float* __restrict__ out) {
    __shared__ float lx[10 * C_];
    __shared__ float red[10 * 8];
    int tid = threadIdx.x, j = blockIdx.x;
    for (int i = tid; i < 10 * C_; i += 256) lx[i] = xave[i];
    __syncthreads();
    float p[10];
#pragma unroll
    for (int b = 0; b < 10; ++b) p[b] = 0.f;
    for (int c = tid; c < C_; c += 256) {
        float wv = Wsw[(size_t)j * C_ + c];
#pragma unroll
        for (int b = 0; b < 10; ++b) p[b] += wv * lx[b * C_ + c];
    }
    int lane = tid & 31, wave = tid >> 5;
#pragma unroll
    for (int b = 0; b < 10; ++b) {
        float v = p[b];
        for (int off = 16; off > 0; off >>= 1) v += __shfl_down(v, off, 32);
        if (lane == 0) red[b * 8 + wave] = v;
    }
    __syncthreads();
    if (tid < 10) {
        float s = 0.f;
#pragma unroll
        for (int w2 = 0; w2 < 8; ++w2) s += red[tid * 8 + w2];  // fixed order: deterministic
        out[5000 + tid * 2000 + j] = s;
    }
}

// ---------------------------------------------------------------------------
// K3: main GEMM  acc[ky][m][n] = x_cat[m,:] . W[n,:]  via V_WMMA_F32_16X16X4_F32.
// Grid (63 n-blocks, 14 k-splits). W tiles DMA'd to LDS by the Tensor Data Mover
// (double-buffered, TENSORcnt), WMMA consumes from LDS while next DMA is in flight.
__global__ __launch_bounds__(256) void gemv_wmma_kernel(const float* __restrict__ Wc,
                                                        const float* __restrict__ xc,
                                                        float* __restrict__ acc) {
    __shared__ float lw[2][16 * LSTR];   // W tile (N=16 rows x K=256), TDM-padded stride 260
    __shared__ float lxs[2][5 * LSTR];   // x_cat tile, 5 real rows
    __shared__ float lzero[LSTR];        // shared zero row for A rows 5..15 (broadcast read)
    __shared__ float red[8 * 8 * 32];    // per-wave C tiles for ordered reduction

    const int tid  = threadIdx.x;
    const int n0   = blockIdx.x * 16;
    const int kb   = blockIdx.y * (TILES_PER_WG * KTILE);
    const int wave = tid >> 5;
    const int lane = tid & 31;
    const int rowm = tid & 15;
    const int half = (tid >> 4) & 1;

    for (int i = tid; i < LSTR; i += 256) lzero[i] = 0.f;

    v8f cacc = {0.f, 0.f, 0.f, 0.f, 0.f, 0.f, 0.f, 0.f};
    const int rem = (NCLS - n0 < 16) ? (NCLS - n0) : 16;   // valid W rows in this n-block

#if USE_TDM
    // ---- Tensor DMA descriptors (CDNA5 ISA 8.3/8.4). Group1 is loop-invariant. ----
    // word0: data_size=4B (2<<16) | pad_enable (1<<20) | pad_interval=256dw (7<<22)
    //        | pad_amount=4dw (3<<25)  -> DMA writes rows at stride LSTR=260 floats
    const int g1w0 = (2 << 16) | (1 << 20) | (7 << 22) | (3 << 25);
    i32x8 g1W = { g1w0,
                  (int)((unsigned)(KTOT & 0xFFFF) << 16),        // tensor_dim0 lo16
                  (int)(((KTOT >> 16) & 0xFFFF) | ((unsigned)rem << 16)), // dim0 hi | dim1 lo
                  (int)((unsigned)KTILE << 16),                  // dim1 hi=0 | tile_dim0
                  16,                                            // tile_dim1=16, tile_dim2=0
                  KTOT, 0, 0 };                                  // tensor_dim0_stride
    i32x8 g1X = g1W;
    g1X[2] = (int)(((KTOT >> 16) & 0xFFFF) | (5u << 16));        // tensor_dim1 = 5
    g1X[4] = 5;                                                  // tile_dim1  = 5
    const i32x4 gz4 = {0, 0, 0, 0};
    const i32x8 gz8 = {0, 0, 0, 0, 0, 0, 0, 0};

    const unsigned long long gaW0 = (unsigned long long)(size_t)(Wc + (size_t)n0 * KTOT + kb);
    const unsigned long long gaX0 = (unsigned long long)(size_t)(xc + kb);

    if (wave == 0) {   // wave-uniform branch; TDM ignores EXEC
        // issue tile 0 into buffer 0
        unsigned long long ga = gaW0;
        u32x4 g0 = { 1u, (unsigned)(size_t)&lw[0][0], (unsigned)ga,
                     (unsigned)((ga >> 32) & 0x1FFFFFFu) | (2u << 30) };
        __builtin_amdgcn_tensor_load_to_lds(g0, g1W, gz4, gz4, gz8, 0);
        ga = gaX0;
        u32x4 g0x = { 1u, (unsigned)(size_t)&lxs[0][0], (unsigned)ga,
                      (unsigned)((ga >> 32) & 0x1FFFFFFu) | (2u << 30) };
        __builtin_amdgcn_tensor_load_to_lds(g0x, g1X, gz4, gz4, gz8, 0);
    }
    for (int t = 0; t < TILES_PER_WG; ++t) {
        const int cur = t & 1;
        if (wave == 0) __builtin_amdgcn_s_wait_tensorcnt(0);  // tile t landed in LDS
        __syncthreads();                                      // publish; prev buffer free
        if (wave == 0 && (t + 1) < TILES_PER_WG) {            // DMA tile t+1 under compute
            const int nxt = (t + 1) & 1;
            unsigned long long ga = gaW0 + (size_t)(t + 1) * KTILE * 4;
            u32x4 g0 = { 1u, (unsigned)(size_t)&lw[nxt][0], (unsigned)ga,
                         (unsigned)((ga >> 32) & 0x1FFFFFFu) | (2u << 30) };
            __builtin_amdgcn_tensor_load_to_lds(g0, g1W, gz4, gz4, gz8, 0);
            ga = gaX0 + (size_t)(t + 1) * KTILE * 4;
            u32x4 g0x = { 1u, (unsigned)(size_t)&lxs[nxt][0], (unsigned)ga,
                          (unsigned)((ga >> 32) & 0x1FFFFFFu) | (2u << 30) };
            __builtin_amdgcn_tensor_load_to_lds(g0x, g1X, gz4, gz4, gz8, 0);
        }
        // --- each wave consumes a disjoint 32-wide K slice of the shared tile ---
        const float* xrow = (rowm < 5) ? &lxs[cur][rowm * LSTR] : lzero;
        const float* wrow = &lw[cur][rowm * LSTR];
        const int kbase = wave * 32;
#pragma unroll
        for (int kk = 0; kk < 32; kk += 4) {
            const int k = kbase + kk + 2 * half;
            v2f a, b;
            a.x = xrow[k];  a.y = xrow[k + 1];   // A[16x4]: lane l -> row l&15
            b.x = wrow[k];  b.y = wrow[k + 1];   // B[4x16]: lane l -> col l&15
            cacc = __builtin_amdgcn_wmma_f32_16x16x4_f32(
                false, a, false, b, (short)0, cacc, false, false);
        }
        __syncthreads();   // tile consumed before its buffer is re-DMA'd
    }
#else
    // ---- fallback: synchronous coalesced float4 loads ----
    const int r   = tid >> 4;
    const int l16 = tid & 15;
    const bool valid = r < rem;
    const float* srow = Wc + (size_t)(n0 + r) * KTOT + kb;
    const float* sxr  = xc + (size_t)r * KTOT + kb;
    for (int t = 0; t < TILES_PER_WG; ++t) {
        __syncthreads();
#pragma unroll
        for (int j = 0; j < 4; ++j) {
            float4 v = make_float4(0.f, 0.f, 0.f, 0.f);
            if (valid) v = ((const float4*)(srow + t * KTILE))[l16 + j * 16];
            *(float4*)(&lw[0][0] + r * LSTR + 4 * (l16 + j * 16)) = v;
        }
        if (r < 5) {
#pragma unroll
            for (int j = 0; j < 4; ++j) {
                float4 v = ((const float4*)(sxr + t * KTILE))[l16 + j * 16];
                *(float4*)(&lxs[0][0] + r * LSTR + 4 * (l16 + j * 16)) = v;
            }
        }
        if (valid && (t + 1 < TILES_PER_WG))
            __builtin_prefetch(srow + (t + 1) * KTILE + 4 * l16, 0, 0);
        __syncthreads();
        const float* xrow = (rowm < 5) ? &lxs[0][rowm * LSTR] : lzero;
        const float* wrow = &lw[0][rowm * LSTR];
        const int kbase = wave * 32;
#pragma unroll
        for (int kk = 0; kk < 32; kk += 4) {
            const int k = kbase + kk + 2 * half;
            v2f a, b;
            a.x = xrow[k];  a.y = xrow[k + 1];
            b.x = wrow[k];  b.y = wrow[k + 1];
            cacc = __builtin_amdgcn_wmma_f32_16x16x4_f32(
                false, a, false, b, (short)0, cacc, false, false);
        }
    }
#endif

    // --- deterministic cross-wave reduction of 16x16 C tiles ---
    __syncthreads();
#pragma unroll
    for (int v = 0; v < 8; ++v) red[(wave * 8 + v) * 32 + lane] = cacc[v];
    __syncthreads();
    if (wave == 0) {
#pragma unroll
        for (int v = 0; v < 8; ++v) {
            float s = 0.f;
#pragma unroll
            for (int w2 = 0; w2 < 8; ++w2) s += red[(w2 * 8 + v) * 32 + lane]; // fixed order
            int M = v + 8 * (lane >> 4);    // C/D layout: VGPR v, lane l -> row v+8*(l>>4)
            int N = lane & 15;
            if (M < 5)
                acc[((size_t)blockIdx.y * 5 + M) * NPAD + n0 + N] = s;  // disjoint: no atomics
        }
    }
}

// ---------------------------------------------------------------------------
// K4: sum the 14 k-split partials (fixed order) + sigmoid -> d_out[0..5000)
__global__ void finalize_kernel(const float* __restrict__ acc, float* __restrict__ out) {
    int gid = blockIdx.x * blockDim.x + threadIdx.x;
    if (gid >= 5 * NCLS) return;
    int m = gid / NCLS, n = gid % NCLS;
    float s = 0.f;
    for (int ky = 0; ky < KSPLIT; ++ky) s += acc[((size_t)ky * 5 + m) * NPAD + n];
    out[gid] = 1.f / (1.f + expf(-s));
}

// ---------------------------------------------------------------------------
extern "C" void kernel_launch(void* const* d_in, const int* in_sizes, int n_in,
                              void* d_out, int out_size, void* d_ws, size_t ws_size,
                              hipStream_t stream) {
    const float* x   = (const float*)d_in[0];   // [10,512,14,14]
    const float* cw  = (const float*)d_in[1];   // [1,512,1,1]
    const float* cb  = (const float*)d_in[2];   // [1]
    const float* Wc  = (const float*)d_in[3];   // [1000, 50176]
    const float* Wsw = (const float*)d_in[4];   // [2000, 512]
    const int*   idx = (const int*)d_in[5];     // [10, 49]
    float* out = (float*)d_out;                 // 5000 + 20000 + 490

    // workspace layout (floats): xc[5*50176] | acc[14*5*1008] | xave[10*512]
    float* xc   = (float*)d_ws;
    float* acc  = xc + 5 * KTOT;
    float* xave = acc + KSPLIT * 5 * NPAD;

    reduce_max_kernel<<<(10 * C_ + 255) / 256, 256, 0, stream>>>(x, xave);
    mask_kernel<<<(10 * P_ + 255) / 256, 256, 0, stream>>>(x, cw, cb, out);
    build_xcat_kernel<<<(5 * KTOT) / 256, 256, 0, stream>>>(x, idx, xc);
    swap_gemm_kernel<<<2000, 256, 0, stream>>>(xave, Wsw, out);
    gemv_wmma_kernel<<<dim3(NB, KSPLIT), 256, 0, stream>>>(Wc, xc, acc);
    finalize_kernel<<<(5 * NCLS + 255) / 256, 256, 0, stream>>>(acc, out);
}